// QValueAggregator_47871705481675
// MI455X (gfx1250) — compile-verified
//
#include <hip/hip_runtime.h>

// Problem constants (match reference: B, L, D)
#define BB 64
#define LL 4096
#define DD 256

typedef __attribute__((ext_vector_type(2))) float v2f;
typedef __attribute__((ext_vector_type(8))) float v8f;

// ---------------------------------------------------------------------------
// Pass 1: values[b,l] = dot(states[b,l,:], w) + bias  via V_WMMA_F32_16X16X4_F32
//
// Per wave: 16 timesteps. D = A x B accumulated over K=256 in chunks of 4.
//   A (16x4)  = w[k0..k0+3] broadcast to all 16 rows
//               (documented layout: VGPR0 = K0 (lanes0-15) / K2 (lanes16-31),
//                                   VGPR1 = K1 / K3  -> float2 @ w[k0 + hi*2])
//   B (4x16)  = states chunk, column n = timestep l0+n
//               (B layout: lanes0-15 hold K0..1, lanes16-31 hold K2..3
//                -> float2 @ states[row l0+n][k0 + hi*2])
// Every column of D is the same dot value per timestep; acc[0] of lanes 0-15
// holds values for timesteps l0..l0+15.
//
// Block = 256 threads = 8 waves = 128 timesteps.  Grid = B*L/128 = 2048.
// Purely HBM-bound: streams 256 MB once (~11 us floor at 23.3 TB/s).
// ---------------------------------------------------------------------------
__global__ __launch_bounds__(256) void qv_values_wmma(
    const float* __restrict__ states,   // [B, L, D]
    const float* __restrict__ w,        // [D]
    const float* __restrict__ bias,     // [1]
    float* __restrict__ out)            // [B, L]
{
    __shared__ float wlds[DD];

    const int tid = threadIdx.x;
    wlds[tid] = w[tid];                 // 256 threads load 256 weights
    __syncthreads();

    const int blk   = blockIdx.x;       // 0 .. B*(L/128)-1
    const int b     = blk >> 5;         // / (L/128 = 32)
    const int chunk = blk & 31;
    const int wave  = tid >> 5;
    const int lane  = tid & 31;
    const int l0    = chunk * 128 + wave * 16;
    const int n     = lane & 15;        // timestep column within tile
    const int hi    = lane >> 4;        // K-half select (0 -> K0..1, 1 -> K2..3)

    const float* srow = states + ((size_t)b * LL + (size_t)(l0 + n)) * DD;

    v8f acc = {};
#pragma unroll 8
    for (int k0 = 0; k0 < DD; k0 += 4) {
        const int koff = k0 + (hi << 1);
        v2f afrag = *(const v2f*)(wlds + koff);   // w broadcast operand (LDS)
        v2f bfrag = *(const v2f*)(srow + koff);   // states operand (global)
        acc = __builtin_amdgcn_wmma_f32_16x16x4_f32(
            /*neg_a=*/false, afrag, /*neg_b=*/false, bfrag,
            /*c_mod=*/(short)0, acc, /*reuse_a=*/false, /*reuse_b=*/false);
    }

    if (lane < 16) {
        out[(size_t)b * LL + (size_t)(l0 + n)] = acc[0] + bias[0];
    }
}

// ---------------------------------------------------------------------------
// Pass 2: in-place reverse cumulative sum along L for each batch row.
// One 1024-thread block per row; thread t owns float4 = elements 4t..4t+3.
// Local suffix sums + Hillis-Steele suffix scan of thread totals in LDS.
// Traffic: 2 MB total -> negligible.
// ---------------------------------------------------------------------------
__global__ __launch_bounds__(1024) void qv_suffix_scan(float* __restrict__ out)
{
    __shared__ float tot[1024];

    const int b = blockIdx.x;
    const int t = threadIdx.x;
    float4* row = (float4*)(out + (size_t)b * LL);

    float4 v = row[t];
    const float s3 = v.w;
    const float s2 = v.z + s3;
    const float s1 = v.y + s2;
    const float s0 = v.x + s1;

    tot[t] = s0;
    __syncthreads();

#pragma unroll
    for (int off = 1; off < 1024; off <<= 1) {
        float x = (t + off < 1024) ? tot[t + off] : 0.0f;
        __syncthreads();
        tot[t] += x;
        __syncthreads();
    }

    // exclusive suffix over later threads
    const float excl = (t < 1023) ? tot[t + 1] : 0.0f;

    float4 q;
    q.x = s0 + excl;
    q.y = s1 + excl;
    q.z = s2 + excl;
    q.w = s3 + excl;
    row[t] = q;
}

// ---------------------------------------------------------------------------
extern "C" void kernel_launch(void* const* d_in, const int* in_sizes, int n_in,
                              void* d_out, int out_size, void* d_ws, size_t ws_size,
                              hipStream_t stream) {
    (void)in_sizes; (void)n_in; (void)d_ws; (void)ws_size; (void)out_size;

    const float* states = (const float*)d_in[0];   // [B, L, D] fp32
    const float* w      = (const float*)d_in[1];   // [D]       fp32
    const float* bias   = (const float*)d_in[2];   // [1]       fp32
    float*       out    = (float*)d_out;           // [B, L]    fp32

    // Pass 1: per-timestep reward values (WMMA f32 path), writes d_out.
    qv_values_wmma<<<dim3((BB * LL) / 128), dim3(256), 0, stream>>>(states, w, bias, out);

    // Pass 2: in-place reverse cumsum per batch row.
    qv_suffix_scan<<<dim3(BB), dim3(1024), 0, stream>>>(out);
}